// SmallGCN_77584289235224
// MI455X (gfx1250) — compile-verified
//
#include <hip/hip_runtime.h>
#include <hip/hip_bf16.h>

#define N_NODES 50000
#define N_EDGES 800000
#define HID 128
#define N_GRAPHS 64
#define N_CLASSES 10

typedef __attribute__((ext_vector_type(2))) float v2f;
typedef __attribute__((ext_vector_type(8))) float v8f;

// ---------------------------------------------------------------------------
// degree / rsqrt helpers
// ---------------------------------------------------------------------------
__global__ void set_one_kernel(float* __restrict__ deg, int n) {
    int i = blockIdx.x * blockDim.x + threadIdx.x;
    if (i < n) deg[i] = 1.0f;  // self-loop
}

__global__ void deg_accum_kernel(const long long* __restrict__ dst,
                                 float* __restrict__ deg, int nEdges) {
    int e = blockIdx.x * blockDim.x + threadIdx.x;
    if (e < nEdges) atomicAdd(&deg[(int)dst[e]], 1.0f);
}

__global__ void rsqrt_ip_kernel(float* __restrict__ d, int n) {
    int i = blockIdx.x * blockDim.x + threadIdx.x;
    if (i < n) d[i] = rsqrtf(d[i]);
}

// ---------------------------------------------------------------------------
// GEMM: Y[M,128] = X[M,128] @ W[128,128]  via V_WMMA_F32_16X16X4_F32
// One wave -> one 16x16 tile, block = 8 waves = 16 rows x 128 cols.
// X row-tile staged in LDS (stride 132 to avoid bank conflicts).
// ---------------------------------------------------------------------------
__global__ __launch_bounds__(256) void gemm_wmma_f32_kernel(
    const float* __restrict__ X, const float* __restrict__ W,
    float* __restrict__ Y) {
    __shared__ float As[16 * 132];

    const int tid  = threadIdx.x;
    const int row0 = blockIdx.x * 16;
    const int wave = tid >> 5;
    const int lane = tid & 31;

    // cooperative load: 16 rows x 128 cols of X -> LDS (512 float4 slots)
    for (int i = tid; i < 512; i += 256) {
        const int r  = i >> 5;   // 0..15
        const int c4 = i & 31;   // 0..31
        const float4 v = *(const float4*)(X + (size_t)(row0 + r) * HID + c4 * 4);
        *(float4*)(As + r * 132 + c4 * 4) = v;
    }
    __syncthreads();

    const int col0 = wave * 16;
    const int half = lane >> 4;   // 0: K=k,k+1   1: K=k+2,k+3
    const int l16  = lane & 15;

    v8f acc = {0.f, 0.f, 0.f, 0.f, 0.f, 0.f, 0.f, 0.f};

    #pragma unroll 4
    for (int k = 0; k < HID; k += 4) {
        const int ka = k + 2 * half;
        v2f a, b;
        // A 16x4 f32 layout: lane<16 row=l16 K={k,k+1}; lane>=16 K={k+2,k+3}
        a.x = As[l16 * 132 + ka];
        a.y = As[l16 * 132 + ka + 1];
        // B 4x16 f32 layout mirrored: VGPR0 = row K=ka across N=col0+l16
        b.x = W[(size_t)ka * HID + col0 + l16];
        b.y = W[(size_t)(ka + 1) * HID + col0 + l16];
        acc = __builtin_amdgcn_wmma_f32_16x16x4_f32(
            /*neg_a=*/false, a, /*neg_b=*/false, b,
            /*c_mod=*/(short)0, acc, /*reuse_a=*/false, /*reuse_b=*/false);
    }

    // C/D layout: VGPR r -> M = r (lanes 0-15) or r+8 (lanes 16-31), N = l16
    #pragma unroll
    for (int r = 0; r < 8; ++r) {
        const int row = row0 + r + 8 * half;
        Y[(size_t)row * HID + col0 + l16] = acc[r];
    }
}

// ---------------------------------------------------------------------------
// edge scatter: agg[dst] += dinv[src]*dinv[dst] * hw[src]   (one wave/edge)
// ---------------------------------------------------------------------------
__global__ __launch_bounds__(256) void edge_scatter_kernel(
    const float* __restrict__ hw, const long long* __restrict__ src,
    const long long* __restrict__ dst, const float* __restrict__ dinv,
    float* __restrict__ agg, int nEdges) {
    const int wave = threadIdx.x >> 5;
    const int lane = threadIdx.x & 31;
    const int e = blockIdx.x * 8 + wave;
    if (e >= nEdges) return;
    const int s = (int)src[e];
    const int d = (int)dst[e];
    const float norm = dinv[s] * dinv[d];
    const float4 v = *(const float4*)(hw + (size_t)s * HID + lane * 4);
    float* p = agg + (size_t)d * HID + lane * 4;
    atomicAdd(p + 0, norm * v.x);
    atomicAdd(p + 1, norm * v.y);
    atomicAdd(p + 2, norm * v.z);
    atomicAdd(p + 3, norm * v.w);
}

// ---------------------------------------------------------------------------
// fused self-loop + bias + ReLU (in place in agg)
// ---------------------------------------------------------------------------
__global__ __launch_bounds__(256) void finalize_relu_kernel(
    float* __restrict__ agg, const float* __restrict__ hw,
    const float* __restrict__ dinv, const float* __restrict__ bias, int n) {
    const int idx = blockIdx.x * blockDim.x + threadIdx.x;  // n * 32 threads
    if (idx >= n * 32) return;
    const int node = idx >> 5;
    const int q    = idx & 31;
    const float dv = dinv[node];
    const float sl = dv * dv;
    const size_t base = (size_t)node * HID + q * 4;
    const float4 a = *(const float4*)(agg + base);
    const float4 h = *(const float4*)(hw + base);
    const float4 b = *(const float4*)(bias + q * 4);
    float4 o;
    o.x = fmaxf(a.x + sl * h.x + b.x, 0.f);
    o.y = fmaxf(a.y + sl * h.y + b.y, 0.f);
    o.z = fmaxf(a.z + sl * h.z + b.z, 0.f);
    o.w = fmaxf(a.w + sl * h.w + b.w, 0.f);
    *(float4*)(agg + base) = o;
}

// ---------------------------------------------------------------------------
// mean-pool accumulation (atomics into [64,128] sums + counts)
// ---------------------------------------------------------------------------
__global__ __launch_bounds__(128) void pool_kernel(
    const float* __restrict__ h, const long long* __restrict__ batch,
    float* __restrict__ pooled, float* __restrict__ cnt) {
    const int n = blockIdx.x;
    const int t = threadIdx.x;  // 0..127
    const int g = (int)batch[n];
    atomicAdd(&pooled[g * HID + t], h[(size_t)n * HID + t]);
    if (t == 0) atomicAdd(&cnt[g], 1.0f);
}

// ---------------------------------------------------------------------------
// final FC: out[g,c] = (pooled[g,:]/max(cnt,1)) @ Wfc + bfc   (640 outputs)
// ---------------------------------------------------------------------------
__global__ __launch_bounds__(640) void fc_kernel(
    const float* __restrict__ pooled, const float* __restrict__ cnt,
    const float* __restrict__ Wfc, const float* __restrict__ bfc,
    float* __restrict__ out) {
    const int tid = threadIdx.x;
    if (tid >= N_GRAPHS * N_CLASSES) return;
    const int g = tid / N_CLASSES;
    const int c = tid % N_CLASSES;
    const float inv = 1.0f / fmaxf(cnt[g], 1.0f);
    float s = 0.f;
    #pragma unroll 8
    for (int k = 0; k < HID; ++k) s += pooled[g * HID + k] * Wfc[k * N_CLASSES + c];
    out[g * N_CLASSES + c] = s * inv + bfc[c];
}

// ---------------------------------------------------------------------------
extern "C" void kernel_launch(void* const* d_in, const int* in_sizes, int n_in,
                              void* d_out, int out_size, void* d_ws, size_t ws_size,
                              hipStream_t stream) {
    (void)in_sizes; (void)n_in; (void)out_size; (void)ws_size;

    const float*     x    = (const float*)d_in[0];
    const long long* edge = (const long long*)d_in[1];   // int64 [2, E]
    const long long* bat  = (const long long*)d_in[2];   // int64 [N]
    const float*     W1   = (const float*)d_in[3];
    const float*     b1   = (const float*)d_in[4];
    const float*     W2   = (const float*)d_in[5];
    const float*     b2   = (const float*)d_in[6];
    const float*     Wfc  = (const float*)d_in[7];
    const float*     bfc  = (const float*)d_in[8];
    float*           out  = (float*)d_out;

    const long long* src = edge;
    const long long* dst = edge + N_EDGES;

    // workspace layout (floats)
    float* ws     = (float*)d_ws;
    float* dinv   = ws;                                  // 50000 (also deg)
    float* bufA   = ws + 51200;                          // 50000*128 (hw)
    float* bufB   = bufA + (size_t)N_NODES * HID;        // 50000*128 (agg/h)
    float* pooled = bufB + (size_t)N_NODES * HID;        // 64*128
    float* cnt    = pooled + N_GRAPHS * HID;             // 64

    const int B = 256;

    // degrees -> dinv_sqrt
    set_one_kernel<<<(N_NODES + B - 1) / B, B, 0, stream>>>(dinv, N_NODES);
    deg_accum_kernel<<<(N_EDGES + B - 1) / B, B, 0, stream>>>(dst, dinv, N_EDGES);
    rsqrt_ip_kernel<<<(N_NODES + B - 1) / B, B, 0, stream>>>(dinv, N_NODES);

    // ---- layer 1 ----
    gemm_wmma_f32_kernel<<<N_NODES / 16, B, 0, stream>>>(x, W1, bufA);
    hipMemsetAsync(bufB, 0, (size_t)N_NODES * HID * sizeof(float), stream);
    edge_scatter_kernel<<<N_EDGES / 8, B, 0, stream>>>(bufA, src, dst, dinv, bufB, N_EDGES);
    finalize_relu_kernel<<<(N_NODES * 32 + B - 1) / B, B, 0, stream>>>(bufB, bufA, dinv, b1, N_NODES);

    // ---- layer 2 ----
    gemm_wmma_f32_kernel<<<N_NODES / 16, B, 0, stream>>>(bufB, W2, bufA);
    hipMemsetAsync(bufB, 0, (size_t)N_NODES * HID * sizeof(float), stream);
    edge_scatter_kernel<<<N_EDGES / 8, B, 0, stream>>>(bufA, src, dst, dinv, bufB, N_EDGES);
    finalize_relu_kernel<<<(N_NODES * 32 + B - 1) / B, B, 0, stream>>>(bufB, bufA, dinv, b2, N_NODES);

    // ---- mean pool + FC ----
    hipMemsetAsync(pooled, 0, (size_t)(N_GRAPHS * HID + N_GRAPHS) * sizeof(float), stream);
    pool_kernel<<<N_NODES, HID, 0, stream>>>(bufB, bat, pooled, cnt);
    fc_kernel<<<1, 640, 0, stream>>>(pooled, cnt, Wfc, bfc, out);
}